// CenterHead_41850161332622
// MI455X (gfx1250) — compile-verified
//
#include <hip/hip_runtime.h>
#include <stdint.h>

// ---------------- problem constants ----------------
#define B_      8
#define C_      2
#define H_      1024
#define W_      1024
#define N_      (H_ * W_)          // 1M per plane
#define K_      500
#define NPLANE  (B_ * C_)          // 16
#define BINS    4096               // key >> 20
#define MCAND   16384              // candidate cap per plane
#define POST_MAX 83
#define NMS_TH   4.0f
#define SCORE_THR 0.1f

// output layout (floats): boxes | scores | clses | valid | keep
#define OFF_BX  0
#define OFF_SC  (B_ * K_ * 9)          // 36000
#define OFF_CL  (OFF_SC + B_ * K_)     // 40000
#define OFF_VA  (OFF_CL + B_ * K_)     // 44000
#define OFF_KP  (OFF_VA + B_ * K_)     // 48000

// streaming-scan config: 4 tiles/block -> grid (64,16) = 1024 WGs for BW saturation
#define SCAN_THREADS 256
#define TILE   4096                // floats per LDS buffer (16KB)
#define ITERS  4
#define CHUNK  (TILE * ITERS)      // 16K elements per block

// ---------------- key mapping (order-preserving float -> u32) ----------------
__device__ __forceinline__ uint32_t f2key(float f) {
    uint32_t u = __float_as_uint(f);
    return (u & 0x80000000u) ? ~u : (u | 0x80000000u);
}
__device__ __forceinline__ float key2f(uint32_t k) {
    return __uint_as_float((k & 0x80000000u) ? (k & 0x7FFFFFFFu) : ~k);
}

// ---------------- CDNA5 async global->LDS (ASYNCcnt-tracked) ----------------
__device__ __forceinline__ void async_load_b128(unsigned lds_off, const void* gaddr) {
    asm volatile("global_load_async_to_lds_b128 %0, %1, off"
                 :: "v"(lds_off), "v"(gaddr) : "memory");
}
__device__ __forceinline__ void wait_async_le4() {
    asm volatile("s_wait_asynccnt 0x4" ::: "memory");
}
__device__ __forceinline__ void wait_async_0() {
    asm volatile("s_wait_asynccnt 0x0" ::: "memory");
}

// issue one 16KB tile: 4 x b128 per thread (256 threads * 4 * 16B = 16KB)
__device__ __forceinline__ void issue_tile(float (*buf)[TILE], int bi,
                                           const float* gbase, int tid) {
#pragma unroll
    for (int q = 0; q < 4; ++q) {
        int fo = q * 1024 + tid * 4;                     // float offset in tile
        unsigned lo = (unsigned)(uintptr_t)&buf[bi][fo]; // low 32 bits = LDS addr
        async_load_b128(lo, (const void*)(gbase + fo));
    }
}

// ---------------- pass 1: per-plane 4096-bin histogram ----------------
__global__ __launch_bounds__(SCAN_THREADS)
void k_hist(const float* __restrict__ heat, uint32_t* __restrict__ ghist) {
    __shared__ float buf[2][TILE];
    __shared__ uint32_t lhist[BINS];
    const int tid = threadIdx.x;
    const int plane = blockIdx.y;
    const float* base = heat + (size_t)plane * N_ + (size_t)blockIdx.x * CHUNK;

    for (int i = tid; i < BINS; i += SCAN_THREADS) lhist[i] = 0;
    issue_tile(buf, 0, base, tid);
    __syncthreads();

    for (int it = 0; it < ITERS; ++it) {
        if (it + 1 < ITERS) {
            issue_tile(buf, (it + 1) & 1, base + (size_t)(it + 1) * TILE, tid);
            if (it + 2 < ITERS)
                __builtin_prefetch(base + (size_t)(it + 2) * TILE + tid * 16, 0, 1);
            wait_async_le4();                 // tile `it` landed (in-order done)
        } else {
            wait_async_0();
        }
        __syncthreads();
        const float* cur = buf[it & 1];
#pragma unroll
        for (int r = 0; r < TILE / SCAN_THREADS; ++r) {
            float v = cur[r * SCAN_THREADS + tid];
            atomicAdd(&lhist[f2key(v) >> 20], 1u);
        }
        __syncthreads();                      // buffer reuse fence
    }
    uint32_t* gp = ghist + (size_t)plane * BINS;
    for (int i = tid; i < BINS; i += SCAN_THREADS)
        if (lhist[i]) atomicAdd(&gp[i], lhist[i]);
}

// ---------------- pass 2: find threshold bin per plane ----------------
__global__ __launch_bounds__(256)
void k_thresh(const uint32_t* __restrict__ ghist, uint32_t* __restrict__ tbin) {
    __shared__ uint32_t segsum[256];
    __shared__ uint32_t segoff[256];
    const int tid = threadIdx.x;
    const int plane = blockIdx.x;
    const uint32_t* h = ghist + (size_t)plane * BINS;
    const int hi = BINS - 1 - tid * 16;       // segment scanned top-down
    uint32_t s = 0;
    for (int i = 0; i < 16; ++i) s += h[hi - i];
    segsum[tid] = s;
    __syncthreads();
    if (tid == 0) {
        uint32_t run = 0;
        for (int i = 0; i < 256; ++i) { segoff[i] = run; run += segsum[i]; }
    }
    __syncthreads();
    uint32_t cum = segoff[tid];
    if (cum < K_ && cum + segsum[tid] >= K_) {        // unique crossing segment
        for (int i = 0; i < 16; ++i) {
            cum += h[hi - i];
            if (cum >= K_) { tbin[plane] = (uint32_t)(hi - i); break; }
        }
    }
}

// ---------------- pass 3: collect candidates >= threshold ----------------
__global__ __launch_bounds__(SCAN_THREADS)
void k_collect(const float* __restrict__ heat, const uint32_t* __restrict__ tbin,
               uint32_t* __restrict__ candCount,
               uint32_t* __restrict__ candKey, uint32_t* __restrict__ candIdx) {
    __shared__ float buf[2][TILE];
    const int tid = threadIdx.x;
    const int plane = blockIdx.y;
    const uint32_t thrKey = tbin[plane] << 20;
    const size_t blockStart = (size_t)blockIdx.x * CHUNK;
    const float* base = heat + (size_t)plane * N_ + blockStart;

    issue_tile(buf, 0, base, tid);
    for (int it = 0; it < ITERS; ++it) {
        if (it + 1 < ITERS) {
            issue_tile(buf, (it + 1) & 1, base + (size_t)(it + 1) * TILE, tid);
            wait_async_le4();
        } else {
            wait_async_0();
        }
        __syncthreads();
        const float* cur = buf[it & 1];
#pragma unroll
        for (int r = 0; r < TILE / SCAN_THREADS; ++r) {
            int li = r * SCAN_THREADS + tid;
            uint32_t key = f2key(cur[li]);
            if (key >= thrKey) {
                uint32_t pos = atomicAdd(&candCount[plane], 1u);
                if (pos < MCAND) {
                    candKey[(size_t)plane * MCAND + pos] = key;
                    candIdx[(size_t)plane * MCAND + pos] =
                        (uint32_t)(blockStart + (size_t)it * TILE + li);
                }
            }
        }
        __syncthreads();
    }
}

// ---------------- pass 4: per-plane bitonic top-K (128KB LDS) ----------------
__global__ __launch_bounds__(1024)
void k_select(const uint32_t* __restrict__ candKey, const uint32_t* __restrict__ candIdx,
              const uint32_t* __restrict__ candCount,
              float* __restrict__ planeScore, uint32_t* __restrict__ planeInd) {
    extern __shared__ uint64_t s_arr[];   // MCAND u64
    const int tid = threadIdx.x;
    const int plane = blockIdx.x;
    uint32_t cnt = candCount[plane];
    if (cnt > MCAND) cnt = MCAND;
    for (int i = tid; i < MCAND; i += 1024) {
        uint64_t v = 0;
        if (i < (int)cnt)
            v = ((uint64_t)candKey[(size_t)plane * MCAND + i] << 32) |
                (uint32_t)~candIdx[(size_t)plane * MCAND + i];   // ties: lower idx first
        s_arr[i] = v;
    }
    __syncthreads();
    for (unsigned k2 = 2; k2 <= MCAND; k2 <<= 1) {
        for (unsigned j = k2 >> 1; j > 0; j >>= 1) {
            for (unsigned i = tid; i < MCAND; i += 1024) {
                unsigned ixj = i ^ j;
                if (ixj > i) {
                    uint64_t a = s_arr[i], b = s_arr[ixj];
                    bool sw = ((i & k2) == 0) ? (a < b) : (a > b);   // descending
                    if (sw) { s_arr[i] = b; s_arr[ixj] = a; }
                }
            }
            __syncthreads();
        }
    }
    for (int r = tid; r < K_; r += 1024) {
        uint64_t c = s_arr[r];
        planeScore[plane * K_ + r] = key2f((uint32_t)(c >> 32));
        planeInd[plane * K_ + r]   = ~(uint32_t)(c & 0xFFFFFFFFu);
    }
}

// ---------------- pass 5: merge classes per batch (top-K over c*K) ----------
__global__ __launch_bounds__(512)
void k_merge(const float* __restrict__ planeScore, const uint32_t* __restrict__ planeInd,
             float* __restrict__ bScore, uint32_t* __restrict__ bInd,
             uint32_t* __restrict__ bCls) {
    __shared__ uint64_t arr[1024];
    const int tid = threadIdx.x;
    const int b = blockIdx.x;
    for (int i = tid; i < 1024; i += 512) {
        uint64_t v = 0;
        if (i < C_ * K_) {
            int c = i / K_, r = i % K_;
            uint32_t key = f2key(planeScore[(b * C_ + c) * K_ + r]);
            v = ((uint64_t)key << 32) | (uint32_t)(1023 - i);    // ties: lower flat idx
        }
        arr[i] = v;
    }
    __syncthreads();
    for (unsigned k2 = 2; k2 <= 1024; k2 <<= 1) {
        for (unsigned j = k2 >> 1; j > 0; j >>= 1) {
            for (unsigned i = tid; i < 1024; i += 512) {
                unsigned ixj = i ^ j;
                if (ixj > i) {
                    uint64_t a = arr[i], bb = arr[ixj];
                    bool sw = ((i & k2) == 0) ? (a < bb) : (a > bb);
                    if (sw) { arr[i] = bb; arr[ixj] = a; }
                }
            }
            __syncthreads();
        }
    }
    if (tid < K_) {
        uint64_t c = arr[tid];
        int flat = 1023 - (int)(uint32_t)(c & 0xFFFFFFFFu);
        int cls = flat / K_, r = flat % K_;
        bScore[b * K_ + tid] = key2f((uint32_t)(c >> 32));
        bInd[b * K_ + tid]   = planeInd[(b * C_ + cls) * K_ + r];
        bCls[b * K_ + tid]   = (uint32_t)cls;
    }
}

// ---------------- pass 6: gather features, assemble boxes, validity ---------
__global__ __launch_bounds__(256)
void k_boxes(const float* __restrict__ reg, const float* __restrict__ rs,
             const float* __restrict__ rc, const float* __restrict__ hei,
             const float* __restrict__ dim, const float* __restrict__ vel,
             const float* __restrict__ bScore, const uint32_t* __restrict__ bInd,
             const uint32_t* __restrict__ bCls, float* __restrict__ out) {
    int t = blockIdx.x * blockDim.x + threadIdx.x;
    if (t >= B_ * K_) return;
    const int b = t / K_;
    const uint32_t ind = bInd[t];
    const float score = bScore[t];
    const float y = (float)(ind / W_), x = (float)(ind % W_);
    const size_t bn = (size_t)b * N_;
    float r0 = reg[bn * 2 + ind], r1 = reg[bn * 2 + N_ + ind];
    float xs = x + r0, ys = y + r1;
    float rot = atan2f(rs[bn + ind], rc[bn + ind]);
    float hh = hei[bn + ind];
    float d0 = dim[bn * 3 + ind], d1 = dim[bn * 3 + N_ + ind], d2 = dim[bn * 3 + 2 * N_ + ind];
    float v0 = vel[bn * 2 + ind], v1 = vel[bn * 2 + N_ + ind];
    float* bx = out + OFF_BX + (size_t)t * 9;
    bx[0] = xs; bx[1] = ys; bx[2] = hh; bx[3] = d0; bx[4] = d1; bx[5] = d2;
    bx[6] = rot; bx[7] = v0; bx[8] = v1;
    bool valid = xs >= -100.f && ys >= -100.f && hh >= -10.f &&
                 xs <= 1124.f && ys <= 1124.f && hh <= 10.f && score > SCORE_THR;
    out[OFF_SC + t] = score;
    out[OFF_CL + t] = (float)bCls[t];
    out[OFF_VA + t] = valid ? 1.f : 0.f;
}

// ---------------- pass 7: circle NMS per batch ------------------------------
__global__ __launch_bounds__(512)
void k_nms(float* __restrict__ out) {
    __shared__ float sx[K_], sy[K_];
    __shared__ uint32_t order[K_], supp[K_], keeps[K_], vArr[K_];
    const int tid = threadIdx.x;
    const int b = blockIdx.x;
    if (tid == 0) {
        // stable argsort(-where(valid, score, -1e9)): scores already descending,
        // so order = valid-first stable partition.
        int c = 0;
        for (int j = 0; j < K_; ++j) if (out[OFF_VA + b * K_ + j] > 0.5f) order[c++] = j;
        for (int j = 0; j < K_; ++j) if (!(out[OFF_VA + b * K_ + j] > 0.5f)) order[c++] = j;
    }
    __syncthreads();
    if (tid < K_) {
        uint32_t o = order[tid];
        sx[tid] = out[OFF_BX + (size_t)(b * K_ + o) * 9 + 0];
        sy[tid] = out[OFF_BX + (size_t)(b * K_ + o) * 9 + 1];
        vArr[tid] = out[OFF_VA + b * K_ + o] > 0.5f ? 1u : 0u;
        supp[tid] = 0;
    }
    for (int i = 0; i < K_; ++i) {
        __syncthreads();
        uint32_t si = supp[i];
        if (!si && tid < K_ && tid > i) {
            float dx = sx[tid] - sx[i], dy = sy[tid] - sy[i];
            if (dx * dx + dy * dy <= NMS_TH) supp[tid] = 1;
        }
    }
    __syncthreads();
    if (tid == 0) {
        int cnt = 0;
        for (int j = 0; j < K_; ++j) {
            int ks = !supp[j];
            cnt += ks;
            keeps[j] = (ks && cnt <= POST_MAX) ? 1u : 0u;
        }
    }
    __syncthreads();
    if (tid < K_)
        out[OFF_KP + b * K_ + order[tid]] = (keeps[tid] && vArr[tid]) ? 1.f : 0.f;
}

// ---------------- host side ----------------
extern "C" void kernel_launch(void* const* d_in, const int* in_sizes, int n_in,
                              void* d_out, int out_size, void* d_ws, size_t ws_size,
                              hipStream_t stream) {
    const float* heat = (const float*)d_in[0];
    const float* reg  = (const float*)d_in[1];
    const float* rs   = (const float*)d_in[2];
    const float* rc   = (const float*)d_in[3];
    const float* hei  = (const float*)d_in[4];
    const float* dim  = (const float*)d_in[5];
    const float* vel  = (const float*)d_in[6];
    float* out = (float*)d_out;

    // workspace carve-up (256B aligned)
    size_t off = 0;
    auto carve = [&](size_t bytes) { size_t o = off; off = (off + bytes + 255) & ~(size_t)255; return o; };
    const size_t histOff   = carve((size_t)NPLANE * BINS * 4);
    const size_t tbinOff   = carve((size_t)NPLANE * 4);
    const size_t ccntOff   = carve((size_t)NPLANE * 4);
    const size_t ckeyOff   = carve((size_t)NPLANE * MCAND * 4);
    const size_t cidxOff   = carve((size_t)NPLANE * MCAND * 4);
    const size_t pscOff    = carve((size_t)NPLANE * K_ * 4);
    const size_t pinOff    = carve((size_t)NPLANE * K_ * 4);
    const size_t bscOff    = carve((size_t)B_ * K_ * 4);
    const size_t bindOff   = carve((size_t)B_ * K_ * 4);
    const size_t bclsOff   = carve((size_t)B_ * K_ * 4);
    if (ws_size < off) return;

    char* ws = (char*)d_ws;
    uint32_t* ghist     = (uint32_t*)(ws + histOff);
    uint32_t* tbin      = (uint32_t*)(ws + tbinOff);
    uint32_t* candCount = (uint32_t*)(ws + ccntOff);
    uint32_t* candKey   = (uint32_t*)(ws + ckeyOff);
    uint32_t* candIdx   = (uint32_t*)(ws + cidxOff);
    float*    pScore    = (float*)(ws + pscOff);
    uint32_t* pInd      = (uint32_t*)(ws + pinOff);
    float*    bScore    = (float*)(ws + bscOff);
    uint32_t* bInd      = (uint32_t*)(ws + bindOff);
    uint32_t* bCls      = (uint32_t*)(ws + bclsOff);

    hipMemsetAsync(ghist, 0, (size_t)NPLANE * BINS * 4, stream);
    hipMemsetAsync(candCount, 0, (size_t)NPLANE * 4, stream);

    dim3 scanGrid(N_ / CHUNK, NPLANE);   // (64, 16) = 1024 workgroups
    k_hist<<<scanGrid, SCAN_THREADS, 0, stream>>>(heat, ghist);
    k_thresh<<<NPLANE, 256, 0, stream>>>(ghist, tbin);
    k_collect<<<scanGrid, SCAN_THREADS, 0, stream>>>(heat, tbin, candCount, candKey, candIdx);
    k_select<<<NPLANE, 1024, (size_t)MCAND * 8, stream>>>(candKey, candIdx, candCount, pScore, pInd);
    k_merge<<<B_, 512, 0, stream>>>(pScore, pInd, bScore, bInd, bCls);
    k_boxes<<<(B_ * K_ + 255) / 256, 256, 0, stream>>>(reg, rs, rc, hei, dim, vel,
                                                       bScore, bInd, bCls, out);
    k_nms<<<B_, 512, 0, stream>>>(out);
    (void)in_sizes; (void)n_in; (void)out_size;
}